// RXGate2_549755814188
// MI455X (gfx1250) — compile-verified
//
#include <hip/hip_runtime.h>
#include <hip/hip_bf16.h>
#include <stdint.h>

// Problem constants (D=2, N=14, INDEX=6, BATCH=1024)
#define BATCHV 1024          // floats per state-vector row
#define PAIRS  8192          // LEFT(64) * RIGHT(128) row pairs
#define NPAIR  4             // pairs per block (pipeline depth)

typedef float v4f __attribute__((ext_vector_type(4)));

__global__ __launch_bounds__(256)
void rx_gate_async_kernel(const float* __restrict__ xr,
                          const float* __restrict__ xi,
                          const float* __restrict__ ang,
                          float* __restrict__ out)
{
    // 2 pipeline stages x 4 logical rows (x0r,x0i,x1r,x1i) x 1024 floats = 32KB
    __shared__ __align__(16) float lds[2][4][BATCHV];

    const int tid = threadIdx.x;
    const int col = tid << 2;                         // 4 batch columns / thread
    const unsigned ldsBase = (unsigned)(uintptr_t)&lds[0][0][0];

    // ---- truncated Taylor gate coefficients (CUTOFF = 10) ----
    const double t  = 0.5 * (double)ang[0];
    const double t2 = t * t;
    const double cD = 1.0 + t2*(-1.0/2.0 + t2*(1.0/24.0 + t2*(-1.0/720.0
                      + t2*(1.0/40320.0 - t2*(1.0/3628800.0)))));
    const double sD = t*(1.0 + t2*(-1.0/6.0 + t2*(1.0/120.0 + t2*(-1.0/5040.0
                      + t2*(1.0/362880.0)))));
    const float c = (float)cD;
    const float s = (float)sD;

    const int pair0 = blockIdx.x * NPAIR;

    // Issue the 4 async b128 copies (global -> LDS) for pair p into buffer buf.
    // Each lane moves exactly the 16B it will later consume -> wave/thread
    // private LDS regions, no workgroup barrier required.
    auto issue = [&](int p, int buf) {
        const int row0 = ((p >> 7) << 8) + (p & 127);     // l*256 + r
        const float* g0 = xr + (size_t)row0 * BATCHV + col;
        const float* g1 = xi + (size_t)row0 * BATCHV + col;
        const float* g2 = xr + (size_t)(row0 + 128) * BATCHV + col;
        const float* g3 = xi + (size_t)(row0 + 128) * BATCHV + col;
        const unsigned b = ldsBase + ((unsigned)(buf * 4 * BATCHV + col) << 2);
        const unsigned stride = BATCHV * 4u;
        asm volatile("global_load_async_to_lds_b128 %0, %1, off"
                     :: "v"(b),              "v"(g0) : "memory");
        asm volatile("global_load_async_to_lds_b128 %0, %1, off"
                     :: "v"(b + stride),     "v"(g1) : "memory");
        asm volatile("global_load_async_to_lds_b128 %0, %1, off"
                     :: "v"(b + 2 * stride), "v"(g2) : "memory");
        asm volatile("global_load_async_to_lds_b128 %0, %1, off"
                     :: "v"(b + 3 * stride), "v"(g3) : "memory");
    };

    issue(pair0, 0);   // prologue: stage 0 in flight

    #pragma unroll
    for (int i = 0; i < NPAIR; ++i) {
        const int p   = pair0 + i;
        const int buf = i & 1;

        if (i + 1 < NPAIR) {
            // prior ds_loads into buf^1 already consumed; be explicit anyway
            asm volatile("s_wait_dscnt 0" ::: "memory");
            issue(p + 1, buf ^ 1);                       // prefetch next stage
            // 8 outstanding, per-wave in-order completion: <=4 means stage i done
            asm volatile("s_wait_asynccnt 4" ::: "memory");
        } else {
            asm volatile("s_wait_asynccnt 0" ::: "memory");
        }

        const v4f x0r = *(const v4f*)&lds[buf][0][col];
        const v4f x0i = *(const v4f*)&lds[buf][1][col];
        const v4f x1r = *(const v4f*)&lds[buf][2][col];
        const v4f x1i = *(const v4f*)&lds[buf][3][col];

        // out0 = c*x0 - i*s*x1 ; out1 = -i*s*x0 + c*x1
        const v4f o0r = c * x0r + s * x1i;
        const v4f o0i = c * x0i - s * x1r;
        const v4f o1r = s * x0i + c * x1r;
        const v4f o1i = c * x1i - s * x0r;

        // interleave to complex64 layout, 2x 16B per output row
        v4f out0a = { o0r.x, o0i.x, o0r.y, o0i.y };
        v4f out0b = { o0r.z, o0i.z, o0r.w, o0i.w };
        v4f out1a = { o1r.x, o1i.x, o1r.y, o1i.y };
        v4f out1b = { o1r.z, o1i.z, o1r.w, o1i.w };

        const int row0 = ((p >> 7) << 8) + (p & 127);
        float* p0 = out + ((size_t)row0 * BATCHV + col) * 2;
        float* p1 = out + ((size_t)(row0 + 128) * BATCHV + col) * 2;
        __builtin_nontemporal_store(out0a, (v4f*)p0);
        __builtin_nontemporal_store(out0b, (v4f*)(p0 + 4));
        __builtin_nontemporal_store(out1a, (v4f*)p1);
        __builtin_nontemporal_store(out1b, (v4f*)(p1 + 4));
    }
}

extern "C" void kernel_launch(void* const* d_in, const int* in_sizes, int n_in,
                              void* d_out, int out_size, void* d_ws, size_t ws_size,
                              hipStream_t stream) {
    const float* x_real = (const float*)d_in[0];   // [16384, 1024] f32
    const float* x_imag = (const float*)d_in[1];   // [16384, 1024] f32
    const float* angle  = (const float*)d_in[2];   // [1] f32
    float* out = (float*)d_out;                    // [16384, 1024] complex64 (interleaved)

    dim3 grid(PAIRS / NPAIR);   // 2048 blocks
    dim3 block(256);            // 8 wave32 waves
    rx_gate_async_kernel<<<grid, block, 0, stream>>>(x_real, x_imag, angle, out);
}